// MoEMamba_50362786512977
// MI455X (gfx1250) — compile-verified
//
#include <hip/hip_runtime.h>
#include <math.h>

// ---------------------------------------------------------------------------
// MoE-Mamba ViT forward for MI455X (gfx1250, wave32).
// All GEMMs run on V_WMMA_F32_16X16X4_F32 (fp32 matrix pipe, exact-precision
// path; model is L2-resident so low-precision WMMA buys nothing here).
// ---------------------------------------------------------------------------

typedef float v2f __attribute__((ext_vector_type(2)));
typedef float v8f __attribute__((ext_vector_type(8)));

#define ACT_NONE     0
#define ACT_SILU     1
#define ACT_GELU     2
#define ACT_SOFTPLUS 3

#define IMG     224
#define PATCH   16
#define EMBED   128
#define NEXP    5
#define DSTATE  16
#define DI      256      // EMBED * EXPAND
#define DTR     8        // EMBED / 16
#define NPATCH  196
#define BATCH   16
#define L_INJ   196
#define L_TOK   197

__device__ __forceinline__ float act_apply(float v, int act) {
  switch (act) {
    case ACT_SILU:     return v / (1.f + expf(-v));
    case ACT_GELU:     return 0.5f * v * (1.f + erff(v * 0.70710678118654752f));
    case ACT_SOFTPLUS: return (v > 20.f) ? v : log1pf(expf(v));
    default:           return v;
  }
}

// ---------------------------------------------------------------------------
// WMMA GEMM: C[M][N] = act(A[M][K] * W[N][K]^T + bias[N])
// One wave computes one 16x16 tile of C; block = 4 waves = 16 rows x 64 cols.
// Requires M % 16 == 0, K % 4 == 0, lda/ldw even (8B-aligned vector loads) —
// true at every call site here.
// A fragment (16x4 f32): lane = {M = lane&15, Kpair = lane>>4}, 2 VGPRs.
// B fragment (4x16 f32): lane = {N = lane&15, Kpair = lane>>4}, 2 VGPRs.
// C/D (16x16 f32, 8 VGPRs): VGPR v -> M = v + 8*(lane>>4), N = lane&15.
// Out-of-range columns: loads are clamped in-bounds (row N-1) and the
// fragment is zeroed with a lane mask multiply — no exec-mask branches.
// ---------------------------------------------------------------------------
__global__ void gemm_wmma_f32_kernel(const float* __restrict__ A, int lda,
                                     const float* __restrict__ W, int ldw,
                                     const float* __restrict__ bias,
                                     float* __restrict__ C, int ldc,
                                     int M, int N, int K, int act) {
  const int lane = threadIdx.x & 31;
  const int wave = threadIdx.x >> 5;
  const int row0 = blockIdx.x << 4;
  const int col0 = (blockIdx.y << 6) + (wave << 4);
  const int nn   = lane & 15;
  const int kg   = lane >> 4;          // which K pair (0 -> K0,K1 ; 1 -> K2,K3)
  const int col  = col0 + nn;
  const bool colok = (col < N);
  const int colc = colok ? col : (N - 1);     // clamped, always in-bounds
  const float bmask = colok ? 1.f : 0.f;

  const float* arow = A + (size_t)(row0 + nn) * lda + (kg << 1);
  const float* wrow = W + (size_t)colc * ldw + (kg << 1);

  v8f acc = {};
#pragma unroll 4
  for (int k0 = 0; k0 < K; k0 += 4) {
    v2f a = *reinterpret_cast<const v2f*>(arow + k0);   // global_load_b64
    v2f b = *reinterpret_cast<const v2f*>(wrow + k0);   // global_load_b64
    b.x *= bmask;
    b.y *= bmask;
    acc = __builtin_amdgcn_wmma_f32_16x16x4_f32(
        /*neg_a=*/false, a, /*neg_b=*/false, b,
        /*c_mod=*/(short)0, acc, /*reuse_a=*/false, /*reuse_b=*/false);
  }

  if (!colok) return;
  const float bv = bias ? bias[col] : 0.f;
  const int mbase = row0 + (kg << 3);
#pragma unroll
  for (int v = 0; v < 8; ++v) {
    float val = act_apply(acc[v] + bv, act);
    C[(size_t)(mbase + v) * ldc + col] = val;
  }
}

// ---------------------------------------------------------------------------
// im2col for the 5 derived single-channel images (R,G,B,gray,pallor).
// P[b*196 + p][u*16+v] = chan(b, py*16+u, px*16+v); p = py*14+px.
// ---------------------------------------------------------------------------
__global__ void im2col_ch_kernel(const float* __restrict__ x,
                                 float* __restrict__ P, int expert) {
  const int total = BATCH * NPATCH * 256;
  int idx = blockIdx.x * blockDim.x + threadIdx.x;
  if (idx >= total) return;
  int colp = idx & 255;
  int p    = idx >> 8;
  int b    = p / NPATCH;
  int pp   = p - b * NPATCH;
  int py = pp / 14, px = pp - (pp / 14) * 14;
  int u = colp >> 4, v = colp & 15;
  int r = py * PATCH + u, cc = px * PATCH + v;
  size_t pix = ((size_t)b * 3) * IMG * IMG + (size_t)r * IMG + cc;
  float R  = x[pix];
  float G  = x[pix + (size_t)IMG * IMG];
  float Bv = x[pix + (size_t)2 * IMG * IMG];
  float val;
  switch (expert) {
    case 0:  val = R;  break;
    case 1:  val = G;  break;
    case 2:  val = Bv; break;
    case 3:  val = 0.299f * R + 0.587f * G + 0.114f * Bv; break;
    default: val = (R - G) / (R + G + 1e-6f); break;
  }
  P[idx] = val;
}

// im2col for the RGB patch embed: col = c*256 + u*16 + v (matches OIHW weight).
__global__ void im2col_rgb_kernel(const float* __restrict__ x,
                                  float* __restrict__ P) {
  const int total = BATCH * NPATCH * 768;
  int idx = blockIdx.x * blockDim.x + threadIdx.x;
  if (idx >= total) return;
  int colp = idx % 768;
  int p    = idx / 768;
  int b    = p / NPATCH;
  int pp   = p - b * NPATCH;
  int py = pp / 14, px = pp - (pp / 14) * 14;
  int ch = colp >> 8;
  int rem = colp & 255;
  int u = rem >> 4, v = rem & 15;
  int r = py * PATCH + u, cc = px * PATCH + v;
  P[idx] = x[(((size_t)b * 3 + ch) * IMG + r) * IMG + cc];
}

// ---------------------------------------------------------------------------
// LayerNorm over rows of width 128. One block (4 waves) per row.
// ---------------------------------------------------------------------------
__global__ void layernorm128_kernel(const float* __restrict__ X,
                                    const float* __restrict__ g,
                                    const float* __restrict__ bta,
                                    float* __restrict__ Y) {
  __shared__ float red[8];
  int row = blockIdx.x, t = threadIdx.x;
  float v = X[(size_t)row * 128 + t];
  float s = v;
  for (int o = 16; o; o >>= 1) s += __shfl_xor(s, o, 32);
  if ((t & 31) == 0) red[t >> 5] = s;
  __syncthreads();
  float mu = (red[0] + red[1] + red[2] + red[3]) * (1.f / 128.f);
  float d = v - mu;
  float s2 = d * d;
  for (int o = 16; o; o >>= 1) s2 += __shfl_xor(s2, o, 32);
  __syncthreads();
  if ((t & 31) == 0) red[4 + (t >> 5)] = s2;
  __syncthreads();
  float var = (red[4] + red[5] + red[6] + red[7]) * (1.f / 128.f);
  Y[(size_t)row * 128 + t] = d * rsqrtf(var + 1e-5f) * g[t] + bta[t];
}

// ---------------------------------------------------------------------------
// Depthwise causal conv (D_CONV=4) + SiLU over the xm half of xz.
// xz: [M][512] (xm = cols 0..255), xc out: [M][256].
// ---------------------------------------------------------------------------
__global__ void conv1d_silu_kernel(const float* __restrict__ xz,
                                   const float* __restrict__ cw,
                                   const float* __restrict__ cb,
                                   float* __restrict__ xc, int L, int total) {
  int idx = blockIdx.x * blockDim.x + threadIdx.x;
  if (idx >= total) return;
  int c = idx & 255;
  int m = idx >> 8;
  int l = m % L;
  int base = m - l;  // row index of l == 0 for this sequence
  float v = cb[c];
#pragma unroll
  for (int j = 0; j < 4; ++j) {
    int ls = l - 3 + j;
    if (ls >= 0) v += cw[c * 4 + j] * xz[(size_t)(base + ls) * 512 + c];
  }
  xc[idx] = v / (1.f + expf(-v));
}

// ---------------------------------------------------------------------------
// Selective scan. One block per batch element; thread = channel (256).
// 16 states per channel live in registers; B_t/C_t broadcast via LDS.
// xdbl row layout: [dt_raw(8) | B(16) | C(16)] -> offsets 8 and 24.
// ---------------------------------------------------------------------------
__global__ void scan_kernel(const float* __restrict__ xdbl,
                            const float* __restrict__ dtb,
                            const float* __restrict__ xc,
                            const float* __restrict__ A_log,
                            float* __restrict__ ys, int L) {
  __shared__ float Bs[DSTATE], Cs[DSTATE];
  int b = blockIdx.x;
  int c = threadIdx.x;  // 0..255
  float An[DSTATE], h[DSTATE];
#pragma unroll
  for (int n = 0; n < DSTATE; ++n) {
    An[n] = -expf(A_log[c * DSTATE + n]);
    h[n] = 0.f;
  }
  for (int l = 0; l < L; ++l) {
    int row = b * L + l;
    if (threadIdx.x < 32) {
      int n = threadIdx.x & 15;
      if (threadIdx.x < 16) Bs[n] = xdbl[(size_t)row * 40 + 8 + n];
      else                  Cs[n] = xdbl[(size_t)row * 40 + 24 + n];
    }
    __syncthreads();
    float dt = dtb[(size_t)row * 256 + c];
    float u  = xc[(size_t)row * 256 + c];
    float du = dt * u;
    float y = 0.f;
#pragma unroll
    for (int n = 0; n < DSTATE; ++n) {
      h[n] = expf(dt * An[n]) * h[n] + du * Bs[n];
      y += h[n] * Cs[n];
    }
    ys[(size_t)row * 256 + c] = y;
    __syncthreads();
  }
}

// y = (ys + xc*D) * silu(z),  z = xz[:, 256+c]
__global__ void ygate_kernel(const float* __restrict__ ys,
                             const float* __restrict__ xc,
                             const float* __restrict__ Dp,
                             const float* __restrict__ xz,
                             float* __restrict__ yb, int total) {
  int idx = blockIdx.x * blockDim.x + threadIdx.x;
  if (idx >= total) return;
  int c = idx & 255;
  int m = idx >> 8;
  float z = xz[(size_t)m * 512 + 256 + c];
  yb[idx] = (ys[idx] + xc[idx] * Dp[c]) * (z / (1.f + expf(-z)));
}

// softmax over the 5-element alpha vector (single thread; trivially small)
__global__ void softmax_alpha_kernel(const float* __restrict__ alpha,
                                     float* __restrict__ w) {
  if (threadIdx.x == 0 && blockIdx.x == 0) {
    float mx = alpha[0];
    for (int i = 1; i < NEXP; ++i) mx = fmaxf(mx, alpha[i]);
    float s = 0.f, e[NEXP];
    for (int i = 0; i < NEXP; ++i) { e[i] = expf(alpha[i] - mx); s += e[i]; }
    for (int i = 0; i < NEXP; ++i) w[i] = e[i] / s;
  }
}

// per-token gate: softmax over 5 experts, top-2 (first index wins ties),
// renormalized weights written to wexp (zeros for non-selected experts).
__global__ void gate_top2_kernel(const float* __restrict__ gsc,
                                 float* __restrict__ wexp, int M) {
  int m = blockIdx.x * blockDim.x + threadIdx.x;
  if (m >= M) return;
  float s[NEXP];
  float mx = -1e30f;
  for (int e = 0; e < NEXP; ++e) { s[e] = gsc[m * NEXP + e]; mx = fmaxf(mx, s[e]); }
  float sum = 0.f;
  for (int e = 0; e < NEXP; ++e) { s[e] = expf(s[e] - mx); sum += s[e]; }
  float inv = 1.f / sum;
  for (int e = 0; e < NEXP; ++e) s[e] *= inv;
  int i1 = 0;
  for (int e = 1; e < NEXP; ++e) if (s[e] > s[i1]) i1 = e;
  int i2 = (i1 == 0) ? 1 : 0;
  for (int e = 0; e < NEXP; ++e) if (e != i1 && s[e] > s[i2]) i2 = e;
  float denom = fmaxf(s[i1] + s[i2], 1e-9f);
  for (int e = 0; e < NEXP; ++e)
    wexp[m * NEXP + e] = (e == i1) ? s[i1] / denom : ((e == i2) ? s[i2] / denom : 0.f);
}

// out += w5[e] * in   (injection mixture)
__global__ void scale_accum_kernel(float* __restrict__ out,
                                   const float* __restrict__ in,
                                   const float* __restrict__ w5, int e, int total) {
  int idx = blockIdx.x * blockDim.x + threadIdx.x;
  if (idx >= total) return;
  out[idx] += w5[e] * in[idx];
}

// tok += wexp[token][e] * mo   (MoE gather-equivalent accumulation)
__global__ void gate_accum_kernel(float* __restrict__ tok,
                                  const float* __restrict__ mo,
                                  const float* __restrict__ wexp, int e, int total) {
  int idx = blockIdx.x * blockDim.x + threadIdx.x;
  if (idx >= total) return;
  int m = idx >> 7;
  tok[idx] += wexp[m * NEXP + e] * mo[idx];
}

__global__ void add_inplace_kernel(float* __restrict__ out,
                                   const float* __restrict__ in, int total) {
  int idx = blockIdx.x * blockDim.x + threadIdx.x;
  if (idx >= total) return;
  out[idx] += in[idx];
}

// tokens[b][l] = (l<196 ? base+inj : cls) + pos[l]
__global__ void assemble_tokens_kernel(const float* __restrict__ base,
                                       const float* __restrict__ inj,
                                       const float* __restrict__ cls,
                                       const float* __restrict__ pos,
                                       float* __restrict__ tok) {
  const int total = BATCH * L_TOK * 128;
  int idx = blockIdx.x * blockDim.x + threadIdx.x;
  if (idx >= total) return;
  int d = idx & 127;
  int bl = idx >> 7;
  int l = bl % L_TOK;
  int b = bl / L_TOK;
  float v;
  if (l < L_INJ) {
    size_t o = (size_t)(b * L_INJ + l) * 128 + d;
    v = base[o] + inj[o];
  } else {
    v = cls[d];
  }
  tok[idx] = v + pos[l * 128 + d];
}

__global__ void extract_cls_kernel(const float* __restrict__ tn,
                                   float* __restrict__ out) {
  int idx = blockIdx.x * blockDim.x + threadIdx.x;
  if (idx >= BATCH * 128) return;
  int b = idx >> 7, d = idx & 127;
  out[idx] = tn[((size_t)b * L_TOK + (L_TOK - 1)) * 128 + d];
}

// ---------------------------------------------------------------------------
// Host-side orchestration
// ---------------------------------------------------------------------------
struct MambaP {
  const float *in_w, *conv_w, *conv_b, *x_w, *dt_w, *dt_b, *A_log, *Dp, *out_w;
};
struct BlockP {
  const float* gate_w;
  MambaP experts[NEXP];
  const float *n1_g, *n1_b, *n2_g, *n2_b;
  const float *ffn_w1, *ffn_b1, *ffn_w2, *ffn_b2;
};
struct Scratch { float *xz, *xc, *xdbl, *dtb, *ysb, *yb; };

static inline MambaP read_mamba(const float** in, int& c) {
  MambaP m;
  m.in_w = in[c++];  m.conv_w = in[c++]; m.conv_b = in[c++];
  m.x_w  = in[c++];  m.dt_w   = in[c++]; m.dt_b   = in[c++];
  m.A_log = in[c++]; m.Dp     = in[c++]; m.out_w  = in[c++];
  return m;
}

static inline void gemm(hipStream_t st, const float* A, int lda,
                        const float* W, int ldw, const float* bias,
                        float* C, int ldc, int M, int N, int K, int act) {
  dim3 grid(M / 16, (N + 63) / 64);
  gemm_wmma_f32_kernel<<<grid, 128, 0, st>>>(A, lda, W, ldw, bias, C, ldc,
                                             M, N, K, act);
}

static void run_mamba(hipStream_t st, const MambaP& mp, const float* X,
                      int Btot, int L, float* Y, const Scratch& s) {
  const int M = Btot * L;
  const int tot = M * 256;
  gemm(st, X, 128, mp.in_w, 128, nullptr, s.xz, 512, M, 512, 128, ACT_NONE);
  conv1d_silu_kernel<<<(tot + 255) / 256, 256, 0, st>>>(s.xz, mp.conv_w,
                                                        mp.conv_b, s.xc, L, tot);
  gemm(st, s.xc, 256, mp.x_w, 256, nullptr, s.xdbl, 40, M, 40, 256, ACT_NONE);
  gemm(st, s.xdbl, 40, mp.dt_w, 8, mp.dt_b, s.dtb, 256, M, 256, 8, ACT_SOFTPLUS);
  scan_kernel<<<Btot, 256, 0, st>>>(s.xdbl, s.dtb, s.xc, mp.A_log, s.ysb, L);
  ygate_kernel<<<(tot + 255) / 256, 256, 0, st>>>(s.ysb, s.xc, mp.Dp, s.xz,
                                                  s.yb, tot);
  gemm(st, s.yb, 256, mp.out_w, 256, nullptr, Y, 128, M, 128, 256, ACT_NONE);
}

extern "C" void kernel_launch(void* const* d_in, const int* in_sizes, int n_in,
                              void* d_out, int out_size, void* d_ws, size_t ws_size,
                              hipStream_t stream) {
  (void)in_sizes; (void)out_size;
  const float* in[512];
  for (int i = 0; i < n_in && i < 512; ++i) in[i] = (const float*)d_in[i];

  // ---- parse params in setup_inputs() insertion order -----------------------
  int c = 0;
  const float* x = in[c++];
  const float *ch_w[NEXP], *ch_b[NEXP], *tln_g[NEXP], *tln_b[NEXP];
  for (int e = 0; e < NEXP; ++e) ch_w[e] = in[c++];
  for (int e = 0; e < NEXP; ++e) ch_b[e] = in[c++];
  const float* rgb_w = in[c++];
  const float* rgb_b = in[c++];
  for (int e = 0; e < NEXP; ++e) tln_g[e] = in[c++];
  for (int e = 0; e < NEXP; ++e) tln_b[e] = in[c++];
  MambaP inj_m[NEXP];
  for (int e = 0; e < NEXP; ++e) inj_m[e] = read_mamba(in, c);
  const float* alpha   = in[c++];
  const float* iln_g   = in[c++];
  const float* iln_b   = in[c++];
  const float* cls_tok = in[c++];
  const float* pos     = in[c++];
  BlockP blk[4];
  for (int b = 0; b < 4; ++b) {
    blk[b].gate_w = in[c++];
    for (int e = 0; e < NEXP; ++e) blk[b].experts[e] = read_mamba(in, c);
    blk[b].n1_g = in[c++]; blk[b].n1_b = in[c++];
    blk[b].n2_g = in[c++]; blk[b].n2_b = in[c++];
    blk[b].ffn_w1 = in[c++]; blk[b].ffn_b1 = in[c++];
    blk[b].ffn_w2 = in[c++]; blk[b].ffn_b2 = in[c++];
  }
  const float* norm_g = in[c++];
  const float* norm_b = in[c++];
  const float* cls_w1 = in[c++]; const float* cls_b1 = in[c++];
  const float* cls_w2 = in[c++]; const float* cls_b2 = in[c++];
  const float* reg_w1 = in[c++]; const float* reg_b1 = in[c++];
  const float* reg_w2 = in[c++]; const float* reg_b2 = in[c++];

  // ---- workspace layout (~56 MB) -------------------------------------------
  const int M1 = BATCH * L_INJ;  // 3136
  const int M2 = BATCH * L_TOK;  // 3152
  size_t off = 0;
  auto alloc = [&](size_t nfloats) -> float* {
    float* p = (float*)((char*)d_ws + off);
    off += (nfloats * sizeof(float) + 255) & ~(size_t)255;
    return p;
  };
  float* P     = alloc((size_t)M1 * 256);
  float* Prgb  = alloc((size_t)M1 * 768);
  float* perch = alloc((size_t)NEXP * M1 * 128);
  float* base  = alloc((size_t)M1 * 128);
  float* injb  = alloc((size_t)M1 * 128);
  float* tln   = alloc((size_t)M1 * 128);
  float* tok   = alloc((size_t)M2 * 128);
  float* xn    = alloc((size_t)M2 * 128);
  float* hb    = alloc((size_t)M2 * 128);
  float* mo    = alloc((size_t)M2 * 128);
  float* f1    = alloc((size_t)M2 * 256);
  float* gsc   = alloc((size_t)M2 * NEXP);
  float* wexp  = alloc((size_t)M2 * NEXP);
  float* w5    = alloc(8);
  float* clsb  = alloc(16 * 128);
  float* h1    = alloc(16 * 64);
  Scratch s;
  s.xz   = alloc((size_t)M2 * 512);
  s.xc   = alloc((size_t)M2 * 256);
  s.xdbl = alloc((size_t)M2 * 40);
  s.dtb  = alloc((size_t)M2 * 256);
  s.ysb  = alloc((size_t)M2 * 256);
  s.yb   = alloc((size_t)M2 * 256);
  (void)ws_size;

  const int TPB = 256;

  // ---- tokenizer: per-channel patch embeds + tok LN -------------------------
  for (int e = 0; e < NEXP; ++e) {
    int tot = M1 * 256;
    im2col_ch_kernel<<<(tot + TPB - 1) / TPB, TPB, 0, stream>>>(x, P, e);
    float* pc = perch + (size_t)e * M1 * 128;
    gemm(stream, P, 256, ch_w[e], 256, ch_b[e], pc, 128, M1, 128, 256, ACT_NONE);
    layernorm128_kernel<<<M1, 128, 0, stream>>>(pc, tln_g[e], tln_b[e], pc);
  }
  {
    int tot = M1 * 768;
    im2col_rgb_kernel<<<(tot + TPB - 1) / TPB, TPB, 0, stream>>>(x, Prgb);
    gemm(stream, Prgb, 768, rgb_w, 768, rgb_b, base, 128, M1, 128, 768, ACT_NONE);
  }

  // ---- injection: inj = sum_i softmax(alpha)[i] * mamba_i(LN(perch_i)) ------
  softmax_alpha_kernel<<<1, 32, 0, stream>>>(alpha, w5);
  hipMemsetAsync(injb, 0, (size_t)M1 * 128 * sizeof(float), stream);
  for (int e = 0; e < NEXP; ++e) {
    float* pc = perch + (size_t)e * M1 * 128;
    layernorm128_kernel<<<M1, 128, 0, stream>>>(pc, iln_g, iln_b, tln);
    run_mamba(stream, inj_m[e], tln, BATCH, L_INJ, mo, s);
    int tot = M1 * 128;
    scale_accum_kernel<<<(tot + TPB - 1) / TPB, TPB, 0, stream>>>(injb, mo, w5, e, tot);
  }

  // ---- assemble tokens (base+inj, cls appended, +pos) -----------------------
  {
    int tot = M2 * 128;
    assemble_tokens_kernel<<<(tot + TPB - 1) / TPB, TPB, 0, stream>>>(
        base, injb, cls_tok, pos, tok);
  }

  // ---- MoE-Mamba blocks -----------------------------------------------------
  for (int b = 0; b < 4; ++b) {
    const BlockP& bp = blk[b];
    layernorm128_kernel<<<M2, 128, 0, stream>>>(tok, bp.n1_g, bp.n1_b, xn);
    gemm(stream, xn, 128, bp.gate_w, 128, nullptr, gsc, NEXP, M2, NEXP, 128, ACT_NONE);
    gate_top2_kernel<<<(M2 + TPB - 1) / TPB, TPB, 0, stream>>>(gsc, wexp, M2);
    for (int e = 0; e < NEXP; ++e) {
      run_mamba(stream, bp.experts[e], xn, BATCH, L_TOK, mo, s);
      int tot = M2 * 128;
      gate_accum_kernel<<<(tot + TPB - 1) / TPB, TPB, 0, stream>>>(tok, mo, wexp, e, tot);
    }
    layernorm128_kernel<<<M2, 128, 0, stream>>>(tok, bp.n2_g, bp.n2_b, hb);
    gemm(stream, hb, 128, bp.ffn_w1, 128, bp.ffn_b1, f1, 256, M2, 256, 128, ACT_GELU);
    gemm(stream, f1, 256, bp.ffn_w2, 256, bp.ffn_b2, mo, 128, M2, 128, 256, ACT_NONE);
    int tot = M2 * 128;
    add_inplace_kernel<<<(tot + TPB - 1) / TPB, TPB, 0, stream>>>(tok, mo, tot);
  }

  // ---- final norm + heads ---------------------------------------------------
  layernorm128_kernel<<<M2, 128, 0, stream>>>(tok, norm_g, norm_b, xn);
  extract_cls_kernel<<<(BATCH * 128 + TPB - 1) / TPB, TPB, 0, stream>>>(xn, clsb);

  float* out = (float*)d_out;
  gemm(stream, clsb, 128, cls_w1, 128, cls_b1, h1, 64, 16, 64, 128, ACT_GELU);
  gemm(stream, h1, 64, cls_w2, 64, cls_b2, out, 2, 16, 2, 64, ACT_NONE);        // cls: [16][2]
  gemm(stream, clsb, 128, reg_w1, 128, reg_b1, h1, 64, 16, 64, 128, ACT_GELU);
  gemm(stream, h1, 64, reg_w2, 64, reg_b2, out + 32, 1, 16, 1, 64, ACT_NONE);   // reg: [16][1]
}